// TCLoss_71536975282441
// MI455X (gfx1250) — compile-verified
//
#include <hip/hip_runtime.h>
#include <math.h>

// ---------------------------------------------------------------------------
// Total-correlation loss for N=2048, D=64 f32 latents on MI455X (gfx1250).
//   log_qz path     : rank-128 GEMM (v_wmma_f32_16x16x4_f32) + fused online LSE
//   log_qz_product  : per-dim two-pass logsumexp (VALU + v_exp_f32), coalesced on d
// ---------------------------------------------------------------------------

#define DLAT 64      // latent dim
#define KDIM 128     // GEMM K = 2*DLAT  (A' = [z^2 | z], B' = [-0.5*e^{-lv} | m*e^{-lv}])

static constexpr float LOG2PI_F = 1.8378770664093453f;

typedef __attribute__((ext_vector_type(2))) float v2f;
typedef __attribute__((ext_vector_type(8))) float v8f;

// ---------------------------------------------------------------------------
// prep: one block (64 threads) per row r in [0,N). Builds:
//   Ap[r,0:64]=z^2, Ap[r,64:128]=z             (i-side GEMM operand)
//   Bp[r,0:64]=-0.5*e^{-lv}, Bp[r,64:128]=m*e^{-lv}   (j-side GEMM operand)
//   cst[r,d]  = -0.5*(lv + log2pi)             (per-(j,d) additive constant)
//   cj[r]     = sum_d -0.5*(m^2 e^{-lv} + lv + log2pi)  (per-j GEMM bias)
// ---------------------------------------------------------------------------
__global__ void tc_prep_kernel(const float* __restrict__ z,
                               const float* __restrict__ zm,
                               const float* __restrict__ zlv,
                               float* __restrict__ Ap,
                               float* __restrict__ Bp,
                               float* __restrict__ cst,
                               float* __restrict__ cj) {
    const int r = blockIdx.x;
    const int d = threadIdx.x;
    const int idx = r * DLAT + d;

    const float zz = z[idx];
    Ap[r * KDIM + d]        = zz * zz;
    Ap[r * KDIM + DLAT + d] = zz;

    const float lv = zlv[idx];
    const float iv = __expf(-lv);
    const float mm = zm[idx];
    Bp[r * KDIM + d]        = -0.5f * iv;
    Bp[r * KDIM + DLAT + d] = mm * iv;
    cst[idx] = -0.5f * (lv + LOG2PI_F);

    __shared__ float red[DLAT];
    red[d] = -0.5f * (mm * mm * iv + lv + LOG2PI_F);
    __syncthreads();
    for (int off = DLAT / 2; off > 0; off >>= 1) {
        if (d < off) red[d] += red[d + off];
        __syncthreads();
    }
    if (d == 0) cj[r] = red[0];
}

// ---------------------------------------------------------------------------
// gemm_lse: one wave (32 lanes) per 16-row i-tile. For each 16-col j-tile:
// 32x v_wmma_f32_16x16x4_f32 over K=128, accumulator spilled through LDS,
// then online logsumexp over the 16 columns (lanes 0..15 own one row each).
// Never materializes S[N,N].
// ---------------------------------------------------------------------------
__global__ void tc_gemm_lse_kernel(const float* __restrict__ Ap,
                                   const float* __restrict__ Bp,
                                   const float* __restrict__ cj,
                                   float* __restrict__ lseS, int N) {
    __shared__ float tile[16 * 16];

    const int lane = threadIdx.x;          // 0..31 (single wave32)
    const int half = lane >> 4;            // K sub-pair selector per ISA A/B layout
    const int l    = lane & 15;            // row (A) / col (B) index within tile
    const int i0   = blockIdx.x * 16;

    const float* Arow = Ap + (size_t)(i0 + l) * KDIM + half * 2;

    float runMax = -3.0e38f;
    float runSum = 0.0f;

    for (int j0 = 0; j0 < N; j0 += 16) {
        const float* Brow = Bp + (size_t)(j0 + l) * KDIM + half * 2;

        v8f c = {0.f, 0.f, 0.f, 0.f, 0.f, 0.f, 0.f, 0.f};
#pragma unroll 4
        for (int kk = 0; kk < KDIM; kk += 4) {
            v2f a = *(const v2f*)(Arow + kk);   // A[M=l, K=kk+2*half .. +1]
            v2f b = *(const v2f*)(Brow + kk);   // B[K=kk+2*half .. +1, N=l]
            c = __builtin_amdgcn_wmma_f32_16x16x4_f32(
                    /*neg_a=*/false, a, /*neg_b=*/false, b,
                    /*c_mod=*/(short)0, c, /*reuse_a=*/false, /*reuse_b=*/false);
        }

        // C/D layout: lane<16 -> rows 0..7 of col=lane; lane>=16 -> rows 8..15.
#pragma unroll
        for (int v = 0; v < 8; ++v)
            tile[(half * 8 + v) * 16 + l] = c[v];
        __syncthreads();

        if (lane < 16) {
            const int r = lane;
            float tmax = -3.0e38f;
#pragma unroll
            for (int cc = 0; cc < 16; ++cc) {
                float val = tile[r * 16 + cc] + cj[j0 + cc];
                tmax = fmaxf(tmax, val);
            }
            const float nm = fmaxf(runMax, tmax);
            float s = runSum * __expf(runMax - nm);
#pragma unroll
            for (int cc = 0; cc < 16; ++cc) {
                float val = tile[r * 16 + cc] + cj[j0 + cc];
                s += __expf(val - nm);
            }
            runMax = nm;
            runSum = s;
        }
        __syncthreads();
    }

    if (lane < 16) lseS[i0 + lane] = runMax + __logf(runSum);
}

// ---------------------------------------------------------------------------
// perdim_lse: block = 64 d-lanes x 4 rows. Per (i,d): two-pass logsumexp over
// all j (max sweep, then sum-of-exp sweep), then LDS-reduce over d -> sumL[i].
// Loads along d are fully coalesced; working set (~1.5 MB) lives in L2/WGP$.
// ---------------------------------------------------------------------------
__global__ void tc_perdim_lse_kernel(const float* __restrict__ z,
                                     const float* __restrict__ zm,
                                     const float* __restrict__ Bp,   // -0.5*e^{-lv} in [ ,0:64)
                                     const float* __restrict__ cst,
                                     float* __restrict__ sumL, int N) {
    const int d    = threadIdx.x & (DLAT - 1);
    const int isub = threadIdx.x >> 6;
    const int i    = blockIdx.x * 4 + isub;

    const float zi = z[i * DLAT + d];

    float vmax = -3.0e38f;
#pragma unroll 4
    for (int j = 0; j < N; ++j) {
        float diff = zi - zm[j * DLAT + d];
        float val  = fmaf(diff * diff, Bp[(size_t)j * KDIM + d], cst[j * DLAT + d]);
        vmax = fmaxf(vmax, val);
    }

    float s = 0.0f;
#pragma unroll 4
    for (int j = 0; j < N; ++j) {
        float diff = zi - zm[j * DLAT + d];
        float val  = fmaf(diff * diff, Bp[(size_t)j * KDIM + d], cst[j * DLAT + d]);
        s += __expf(val - vmax);
    }
    const float L = vmax + __logf(s);

    __shared__ float red[4][DLAT];
    red[isub][d] = L;
    __syncthreads();
    for (int off = DLAT / 2; off > 0; off >>= 1) {
        if (d < off) red[isub][d] += red[isub][d + off];
        __syncthreads();
    }
    if (d == 0) sumL[i] = red[isub][0];
}

// ---------------------------------------------------------------------------
// finalize: mean_i( sumL[i] - lseS[i] )
// ---------------------------------------------------------------------------
__global__ void tc_finalize_kernel(const float* __restrict__ sumL,
                                   const float* __restrict__ lseS,
                                   float* __restrict__ out, int N) {
    __shared__ float red[256];
    float acc = 0.0f;
    for (int i = threadIdx.x; i < N; i += 256) acc += sumL[i] - lseS[i];
    red[threadIdx.x] = acc;
    __syncthreads();
    for (int off = 128; off > 0; off >>= 1) {
        if ((int)threadIdx.x < off) red[threadIdx.x] += red[threadIdx.x + off];
        __syncthreads();
    }
    if (threadIdx.x == 0) out[0] = red[0] / (float)N;
}

extern "C" void kernel_launch(void* const* d_in, const int* in_sizes, int n_in,
                              void* d_out, int out_size, void* d_ws, size_t ws_size,
                              hipStream_t stream) {
    (void)n_in; (void)out_size; (void)ws_size;

    const float* z   = (const float*)d_in[0];
    const float* zm  = (const float*)d_in[1];
    const float* zlv = (const float*)d_in[2];
    const int N = in_sizes[0] / DLAT;   // 2048 for the reference

    // Workspace layout (floats): Ap[N*128] | Bp[N*128] | cst[N*64] | cj[N] | lseS[N] | sumL[N]
    float* ws   = (float*)d_ws;
    float* Ap   = ws;
    float* Bp   = Ap  + (size_t)N * KDIM;
    float* cst  = Bp  + (size_t)N * KDIM;
    float* cj   = cst + (size_t)N * DLAT;
    float* lseS = cj  + N;
    float* sumL = lseS + N;

    tc_prep_kernel<<<N, DLAT, 0, stream>>>(z, zm, zlv, Ap, Bp, cst, cj);
    tc_gemm_lse_kernel<<<N / 16, 32, 0, stream>>>(Ap, Bp, cj, lseS, N);
    tc_perdim_lse_kernel<<<N / 4, 256, 0, stream>>>(z, zm, Bp, cst, sumL, N);
    tc_finalize_kernel<<<1, 256, 0, stream>>>(sumL, lseS, (float*)d_out, N);
}